// HybridQuanvLayerTQ_65266323030225
// MI455X (gfx1250) — compile-verified
//
#include <hip/hip_runtime.h>
#include <hip/hip_bf16.h>

// out[b,w] = enc_b^T A_w enc_b, enc_b = real product state from the RY encoder,
// A_w = Re(U_ansatz^dag Z_w U_ansatz) fixed per call (symmetric!).
// Symmetry-folded Gram formulation: K runs over the 136 pairs (j<=k), padded
// to 144; Bmat[K][n] = (j==k ? 1 : 2) * A_n[j][k] for n<4, else 0.
// GEMM (Bx144)x(144x16) on v_wmma_f32_16x16x4_f32: 36 chunks per 16-row tile.

typedef float v2f __attribute__((ext_vector_type(2)));
typedef float v8f __attribute__((ext_vector_type(8)));

#define DIM 16
#define NPAIR 136      // 16*17/2
#define NK 144         // padded to multiple of 4
#define NCHUNK 36      // NK / 4

// K -> (j,k) pair table, j<=k, row-major over the upper triangle. Built as a
// constexpr aggregate so constant-index reads (after full unroll) fold to
// literals — a constexpr *function* with a loop was emitted as runtime SALU
// loops at every call site (round-2 salu blowup), tables fold reliably.
struct PairTab {
    int j[NK];
    int k[NK];
    constexpr PairTab() : j{}, k{} {
        int K = 0;
        for (int a = 0; a < DIM; ++a)
            for (int b = a; b < DIM; ++b) { j[K] = a; k[K] = b; ++K; }
        for (; K < NK; ++K) { j[K] = 0; k[K] = 0; }
    }
};
constexpr PairTab PT{};

// ---------------- Kernel 1: build triangular-packed Bmat --------------------

__device__ __forceinline__ void apply_ry(float* re, float* im, float th, int w) {
    int m = 8 >> w;                       // wire w -> bit (3-w)
    float c = __cosf(0.5f * th), s = __sinf(0.5f * th);
    for (int i = 0; i < DIM; ++i) {
        if (i & m) continue;
        int j = i | m;
        float r0 = re[i], q0 = im[i], r1 = re[j], q1 = im[j];
        re[i] = c * r0 - s * r1;  im[i] = c * q0 - s * q1;
        re[j] = s * r0 + c * r1;  im[j] = s * q0 + c * q1;
    }
}

__device__ __forceinline__ void apply_rz(float* re, float* im, float th, int w) {
    int m = 8 >> w;
    float c = __cosf(0.5f * th), s = __sinf(0.5f * th);  // e^{-i th/2} = c - i s
    for (int i = 0; i < DIM; ++i) {
        float r = re[i], q = im[i];
        if (i & m) { re[i] = c * r - s * q; im[i] = c * q + s * r; }   // * (c + i s)
        else       { re[i] = c * r + s * q; im[i] = c * q - s * r; }   // * (c - i s)
    }
}

__device__ __forceinline__ void apply_cnot(float* re, float* im, int cw, int tw) {
    int cm = 8 >> cw, tm = 8 >> tw;
    for (int i = 0; i < DIM; ++i) {
        if ((i & cm) && !(i & tm)) {
            int j = i | tm;
            float r = re[i]; re[i] = re[j]; re[j] = r;
            float q = im[i]; im[i] = im[j]; im[j] = q;
        }
    }
}

__global__ void build_quadforms(const float* __restrict__ params,
                                float* __restrict__ Bm /* [NK][16] */) {
    __shared__ float ure[DIM][DIM];
    __shared__ float uim[DIM][DIM];
    int t = threadIdx.x;
    if (t < DIM) {
        // column t of the ansatz unitary: evolve basis state e_t
        float re[DIM], im[DIM];
        for (int i = 0; i < DIM; ++i) { re[i] = (i == t) ? 1.f : 0.f; im[i] = 0.f; }
        const int pa[4] = {0, 1, 2, 3};
        const int pb[4] = {1, 2, 3, 0};
        for (int layer = 0; layer < 2; ++layer) {
            const float* p = params + layer * 7;
            for (int e = 0; e < 4; ++e) {
                int w0 = pa[e], w1 = pb[e];
                apply_ry(re, im, p[0], w0);
                apply_rz(re, im, p[1], w0);
                apply_ry(re, im, p[2], w1);
                apply_rz(re, im, p[3], w1);
                apply_cnot(re, im, w0, w1);
                apply_ry(re, im, p[4], w0);
                apply_rz(re, im, p[5], w0);
                apply_ry(re, im, p[6], w1);
            }
        }
        for (int i = 0; i < DIM; ++i) { ure[i][t] = re[i]; uim[i][t] = im[i]; }
    }
    __syncthreads();
    // Bm[K][n] = (j==k ? 1 : 2) * sum_i z_n(i) Re(conj(U[i][j]) U[i][k]), n<4
    // NK*16 = 2304 entries over 32 threads = 72 each
    for (int e = 0; e < 72; ++e) {
        int idx = t * 72 + e;
        int K = idx >> 4, col = idx & 15;
        float val = 0.f;
        if (K < NPAIR && col < 4) {
            int j = PT.j[K], k = PT.k[K];
            int m = 8 >> col;                 // z_col(i) sign mask
            float acc = 0.f;
            for (int i = 0; i < DIM; ++i) {
                float z = (i & m) ? -1.f : 1.f;
                acc += z * (ure[i][j] * ure[i][k] + uim[i][j] * uim[i][k]);
            }
            val = (j == k) ? acc : 2.f * acc;
        }
        Bm[idx] = val;
    }
}

// ---------------- Kernel 2: batched quadratic forms via WMMA ----------------

__global__ __launch_bounds__(256) void qform_wmma(
        const float* __restrict__ patches,  // [B][4]
        const float* __restrict__ Bm,       // [NK][16] from kernel 1
        float* __restrict__ out) {          // [B][4]
    __shared__ float bmat[NK * 16];         // 9 KB
    int tid = threadIdx.x;
    #pragma unroll
    for (int i = 0; i < (NK * 16) / 256; ++i) {   // 9 loads per thread
        int flat = tid + i * 256;
        bmat[flat] = Bm[flat];
    }
    __syncthreads();

    int lane = tid & 31;
    int wv   = tid >> 5;
    int row  = lane & 15;                       // matrix row / output column index
    int hi   = lane >> 4;                       // half-wave: K offset +2 in operands
    int tileBase = (blockIdx.x * 8 + wv) * 16;  // 16 batch rows per wave
    int b = tileBase + row;

    // encoder: enc[j] = prod_i (bit_i(j) ? sin(x_i/2) : cos(x_i/2)), x = pi*sigmoid
    float4 xv = ((const float4*)patches)[b];
    float xs[4] = {xv.x, xv.y, xv.z, xv.w};
    float c[4], s[4];
    #pragma unroll
    for (int i = 0; i < 4; ++i) {
        float sg = 1.f / (1.f + __expf(-xs[i]));
        float h  = 1.57079632679489662f * sg;   // (pi * sigmoid) / 2
        c[i] = __cosf(h);
        s[i] = __sinf(h);
    }
    float t01[4] = {c[0] * c[1], c[0] * s[1], s[0] * c[1], s[0] * s[1]};
    float t23[4] = {c[2] * c[3], c[2] * s[3], s[2] * c[3], s[2] * s[3]};
    float enc[16];
    #pragma unroll
    for (int j = 0; j < 16; ++j) enc[j] = t01[j >> 2] * t23[j & 3];

    // A operand chunk q: lanes 0-15 carry K = 4q,4q+1; lanes 16-31 K = 4q+2,4q+3
    // g(K) = enc[PT.j[K]] * enc[PT.k[K]] (0 in padding rows) — all indices are
    // compile-time constants under the unroll, hi selects via cndmask.
    v8f acc = {};
    #pragma unroll
    for (int q = 0; q < NCHUNK; ++q) {
        const int K0 = 4 * q;
        const int K2 = 4 * q + 2;
        float g0lo = (K0     < NPAIR) ? enc[PT.j[K0    ]] * enc[PT.k[K0    ]] : 0.f;
        float g1lo = (K0 + 1 < NPAIR) ? enc[PT.j[K0 + 1]] * enc[PT.k[K0 + 1]] : 0.f;
        float g0hi = (K2     < NPAIR) ? enc[PT.j[K2    ]] * enc[PT.k[K2    ]] : 0.f;
        float g1hi = (K2 + 1 < NPAIR) ? enc[PT.j[K2 + 1]] * enc[PT.k[K2 + 1]] : 0.f;
        v2f av = { hi ? g0hi : g0lo, hi ? g1hi : g1lo };
        int Kb = 4 * q + 2 * hi;
        v2f bv = { bmat[(Kb << 4) + row], bmat[((Kb + 1) << 4) + row] };
        acc = __builtin_amdgcn_wmma_f32_16x16x4_f32(
            /*neg_a=*/false, av, /*neg_b=*/false, bv,
            /*c_mod=*/(short)0, acc, /*reuse_a=*/false, /*reuse_b=*/false);
    }

    // D layout: VGPR r, lanes 0-15 -> (M=r, N=lane); lanes 16-31 -> (M=r+8, N=lane-16)
    if (row < 4) {
        #pragma unroll
        for (int r = 0; r < 8; ++r) {
            int M = r + hi * 8;
            out[(tileBase + M) * 4 + row] = acc[r];
        }
    }
}

// ---------------- launch -----------------------------------------------------

extern "C" void kernel_launch(void* const* d_in, const int* in_sizes, int n_in,
                              void* d_out, int out_size, void* d_ws, size_t ws_size,
                              hipStream_t stream) {
    const float* patches = (const float*)d_in[0];   // [B][4] fp32
    const float* params  = (const float*)d_in[1];   // [2][7] fp32
    float* out = (float*)d_out;                     // [B][4] fp32
    float* Bm  = (float*)d_ws;                      // NK*16 = 2304 floats scratch

    int B = in_sizes[0] / 4;

    build_quadforms<<<1, 32, 0, stream>>>(params, Bm);

    int blocks = B / 128;                           // 8 waves x 16 rows per block
    qform_wmma<<<blocks, 256, 0, stream>>>(patches, Bm, out);
}